// PartialDecoder_38671885533516
// MI455X (gfx1250) — compile-verified
//
#include <hip/hip_runtime.h>
#include <hip/hip_bf16.h>

// Problem constants from the reference.
#define B_ 64
#define L_ 32
#define H_ 128
#define O_ 16384
#define C_ 32

typedef __attribute__((ext_vector_type(2))) float v2f;
typedef __attribute__((ext_vector_type(8))) float v8f;
typedef __attribute__((ext_vector_type(4))) unsigned int v4u;
typedef __attribute__((ext_vector_type(4))) int v4i;
typedef __attribute__((ext_vector_type(8))) int v8i;

// Address-space-qualified pointer types for the async global->LDS builtin:
// signature is (v4i addrspace(1)* src, v4i addrspace(3)* dst, imm off, imm cpol)
typedef __attribute__((address_space(1))) v4i g_v4i;
typedef __attribute__((address_space(3))) v4i l_v4i;

// LDS byte offset of a generic pointer into shared memory.
__device__ __forceinline__ unsigned lds_off(const void* p) {
    return (unsigned)(unsigned long long)(__attribute__((address_space(3))) const void*)p;
}

__device__ __forceinline__ void wait_tensorcnt0() {
#if __has_builtin(__builtin_amdgcn_s_wait_tensorcnt)
    __builtin_amdgcn_s_wait_tensorcnt(0);
#else
    asm volatile("s_wait_tensorcnt 0x0" ::: "memory");
#endif
}

__device__ __forceinline__ void wait_asynccnt0() {
#if __has_builtin(__builtin_amdgcn_s_wait_asynccnt)
    __builtin_amdgcn_s_wait_asynccnt(0);
#else
    asm volatile("s_wait_asynccnt 0x0" ::: "memory");
#endif
}

// ---------------------------------------------------------------------------
// Kernel 1: A[b,h] = relu(z @ Wz + bz) @ W1_h     (64 x 128, tiny)
// grid = 8 blocks x 256 threads; block j handles b in [8j, 8j+8)
// ---------------------------------------------------------------------------
__global__ void k_prep_A(const float* __restrict__ z,
                         const float* __restrict__ Wz,
                         const float* __restrict__ bz,
                         const float* __restrict__ W1,
                         float* __restrict__ Aout) {
    __shared__ float pz[8 * H_];
    const int tid = threadIdx.x;
    const int b0  = blockIdx.x * 8;

    for (int i = tid; i < 8 * H_; i += 256) {
        const int b = b0 + (i >> 7);
        const int h = i & (H_ - 1);
        float acc = bz[h];
        #pragma unroll
        for (int l = 0; l < L_; ++l)
            acc += z[b * L_ + l] * Wz[l * H_ + h];
        pz[i] = fmaxf(acc, 0.0f);
    }
    __syncthreads();

    for (int i = tid; i < 8 * H_; i += 256) {
        const int bl = i >> 7;
        const int h  = i & (H_ - 1);
        float acc = 0.0f;
        #pragma unroll 8
        for (int hp = 0; hp < H_; ++hp)
            acc += pz[bl * H_ + hp] * W1[hp * H_ + h];
        Aout[(b0 + bl) * H_ + h] = acc;
    }
}

// ---------------------------------------------------------------------------
// Kernel 2: E'[o,h] = (FE @ W1_f)[o,h] + fb[o]*w1b[h] + b1[h]
// f32 GEMM 16384x128 (K=32) on the matrix pipe: V_WMMA_F32_16X16X4_F32.
// grid = O/16 = 1024 blocks x 256 threads (8 waves); wave w owns the
// 16(o) x 16(h) tile at h0 = 16*w, accumulating 8 K=4 WMMA steps.
// ---------------------------------------------------------------------------
__global__ void k_build_E(const float* __restrict__ FE,
                          const float* __restrict__ fb,
                          const float* __restrict__ W1,
                          const float* __restrict__ b1,
                          float* __restrict__ Ep) {
    const float* __restrict__ W1f = W1 + H_ * H_;          // rows 128..159 (C x H)
    const float* __restrict__ w1b = W1 + (H_ + C_) * H_;   // row 160       (1 x H)

    const int lane = threadIdx.x & 31;
    const int wave = threadIdx.x >> 5;
    const int o0   = blockIdx.x * 16;
    const int h0   = wave * 16;

    const int M  = lane & 15;          // A-matrix row (o), also B/D column (h)
    const int kh = (lane >> 4) << 1;   // K sub-offset: 0 for lanes 0-15, 2 for 16-31

    v8f acc = {};
    const float* __restrict__ arow = FE + (o0 + M) * C_;

    #pragma unroll
    for (int k = 0; k < C_; k += 4) {
        v2f av;
        av.x = arow[k + kh + 0];
        av.y = arow[k + kh + 1];
        v2f bv;
        bv.x = W1f[(k + kh + 0) * H_ + h0 + M];
        bv.y = W1f[(k + kh + 1) * H_ + h0 + M];
        acc = __builtin_amdgcn_wmma_f32_16x16x4_f32(
            false, av, false, bv, (short)0, acc, false, false);
    }

    const int   mhalf = (lane >> 4) * 8;
    const float b1v   = b1[h0 + M];
    const float wbv   = w1b[h0 + M];
    #pragma unroll
    for (int r = 0; r < 8; ++r) {
        const int orow = o0 + mhalf + r;
        Ep[orow * H_ + h0 + M] = acc[r] + fb[orow] * wbv + b1v;
    }
}

// ---------------------------------------------------------------------------
// Kernel 3 (dominant): out[b,o] = sum_h relu(A[b,h] + E'[o,h]) * w2[h] + b2
// grid = O/32 = 512 blocks x 256 threads (8 waves).
// A (contiguous 64x128 f32 tile) is fetched by the Tensor Data Mover;
// the E' tile is fetched with GLOBAL_LOAD_ASYNC_TO_LDS_B128. Lanes index o
// (per-lane E' reads, stride-132 pad keeps 16B alignment); A and w2 reads
// are wave-uniform LDS broadcasts; 8 accumulators per wave give 8x reuse
// of every E' element.
// ---------------------------------------------------------------------------
#define EP_STRIDE 132   // 132*4 = 528 bytes: 16B-aligned rows, banks spread by 4

__global__ void k_main(const float* __restrict__ Ep,
                       const float* __restrict__ Afull,
                       const float* __restrict__ W2,
                       const float* __restrict__ b2,
                       float* __restrict__ out) {
    __shared__ float Elds[32 * EP_STRIDE];
    __shared__ float Alds[B_ * H_];
    __shared__ float w2s[H_];

    const int tid  = threadIdx.x;
    const int lane = tid & 31;
    const int wave = tid >> 5;
    const int o0   = blockIdx.x * 32;

    // ---- A tile: one TDM descriptor issued by wave 0 (32 KB linear 2D tile)
#if __has_builtin(__builtin_amdgcn_tensor_load_to_lds)
    bool tdm_wait = false;
    if (wave == 0) {
        const unsigned lbase = lds_off(&Alds[0]);
        const unsigned long long ga = (unsigned long long)(const void*)Afull;
        v4u g0;
        g0[0] = 1u;                                   // count=1, no gather
        g0[1] = lbase;                                // lds_addr (bytes)
        g0[2] = (unsigned)ga;                         // global_addr[31:0]
        g0[3] = (unsigned)(ga >> 32) | (2u << 30);    // global_addr[56:32] | type=2
        v8i g1;
        g1[0] = (int)(2u << 16);    // wg_mask=0, data_size=2 (4B), no iterate/pad
        g1[1] = (int)(128u << 16);  // tensor_dim0[15:0] = 128
        g1[2] = (int)(64u << 16);   // tensor_dim0 hi=0 | tensor_dim1[15:0] = 64
        g1[3] = (int)(128u << 16);  // tensor_dim1 hi=0 | tile_dim0 = 128
        g1[4] = 64;                 // tile_dim1 = 64, tile_dim2 = 0
        g1[5] = 128;                // tensor_dim0_stride[31:0] = 128
        g1[6] = 0;                  // stride hi | tensor_dim1_stride lo = 0
        g1[7] = 0;
        v4i g2 = {0, 0, 0, 0};
        v4i g3 = {0, 0, 0, 0};
#if __clang_major__ >= 23
        v8i g4 = {0, 0, 0, 0, 0, 0, 0, 0};
        __builtin_amdgcn_tensor_load_to_lds(g0, g1, g2, g3, g4, 0);
#else
        __builtin_amdgcn_tensor_load_to_lds(g0, g1, g2, g3, 0);
#endif
        tdm_wait = true;
    }
#else
    for (int i = tid; i < B_ * H_; i += 256)
        Alds[i] = Afull[i];
#endif

    // ---- E' tile: 32 rows x 512B via per-lane async b128 copies (1024 chunks)
#if __has_builtin(__builtin_amdgcn_global_load_async_to_lds_b128)
    #pragma unroll
    for (int i = tid; i < 32 * 32; i += 256) {
        const int r = i >> 5;         // o-row within tile
        const int c = i & 31;         // 16B chunk within row
        g_v4i* src = (g_v4i*)(Ep + (size_t)(o0 + r) * H_ + c * 4);
        l_v4i* dst = (l_v4i*)(Elds + r * EP_STRIDE + c * 4);
        __builtin_amdgcn_global_load_async_to_lds_b128(src, dst, 0, 0);
    }
    if (tid < H_) w2s[tid] = W2[tid];
    wait_asynccnt0();
#else
    for (int i = tid; i < 32 * H_; i += 256) {
        const int r = i >> 7;
        const int c = i & (H_ - 1);
        Elds[r * EP_STRIDE + c] = Ep[(size_t)(o0 + r) * H_ + c];
    }
    if (tid < H_) w2s[tid] = W2[tid];
#endif

#if __has_builtin(__builtin_amdgcn_tensor_load_to_lds)
    if (tdm_wait) wait_tensorcnt0();
#endif
    __syncthreads();

    const float b2v = b2[0];
    const float* __restrict__ erow = &Elds[lane * EP_STRIDE];

    float acc[8];
    #pragma unroll
    for (int bb = 0; bb < 8; ++bb) acc[bb] = 0.0f;

    #pragma unroll 4
    for (int h = 0; h < H_; ++h) {
        const float e = erow[h];   // per-lane
        const float w = w2s[h];    // uniform broadcast
        #pragma unroll
        for (int bb = 0; bb < 8; ++bb) {
            const float a = Alds[(wave * 8 + bb) * H_ + h];  // uniform broadcast
            acc[bb] += fmaxf(a + e, 0.0f) * w;
        }
    }

    #pragma unroll
    for (int bb = 0; bb < 8; ++bb) {
        const int b = wave * 8 + bb;
        out[(size_t)b * O_ + o0 + lane] = acc[bb] + b2v;
    }
}

// ---------------------------------------------------------------------------
extern "C" void kernel_launch(void* const* d_in, const int* in_sizes, int n_in,
                              void* d_out, int out_size, void* d_ws, size_t ws_size,
                              hipStream_t stream) {
    const float* z  = (const float*)d_in[0];  // (64,32)
    const float* FE = (const float*)d_in[1];  // (16384,32)
    const float* fb = (const float*)d_in[2];  // (16384,1)
    const float* Wz = (const float*)d_in[3];  // (32,128)
    const float* bz = (const float*)d_in[4];  // (128,)
    const float* W1 = (const float*)d_in[5];  // (161,128)
    const float* b1 = (const float*)d_in[6];  // (128,)
    const float* W2 = (const float*)d_in[7];  // (128,1)
    const float* b2 = (const float*)d_in[8];  // (1,)
    float* out = (float*)d_out;               // (64,16384)

    float* Ep    = (float*)d_ws;              // 16384*128 f32 = 8 MB
    float* Afull = Ep + (size_t)O_ * H_;      // 64*128 f32

    k_prep_A<<<8, 256, 0, stream>>>(z, Wz, bz, W1, Afull);
    k_build_E<<<O_ / 16, 256, 0, stream>>>(FE, fb, W1, b1, Ep);
    k_main<<<O_ / 32, 256, 0, stream>>>(Ep, Afull, W2, b2, out);
}